// FourDRNN_91070486544620
// MI455X (gfx1250) — compile-verified
//
#include <hip/hip_runtime.h>
#include <math.h>

// Problem constants (from reference): B=4096, C=200, P=5 (T=P*P=25), H=64, NCLS=16
#define NB   4096
#define NC   200
#define NT   25
#define NH   64
#define NG   256      // 4*H gates
#define NF   1600     // H * T features
#define NCLS 16

typedef __attribute__((ext_vector_type(2))) float v2f;
typedef __attribute__((ext_vector_type(8))) float v8f;

// CDNA5 fp32 WMMA: D(16x16,f32) = A(16x4,f32) * B(4x16,f32) + C
#define WMMA_F32(a, b, c) \
    __builtin_amdgcn_wmma_f32_16x16x4_f32(false, (a), false, (b), (short)0, (c), false, false)

__device__ __forceinline__ float sigf(float x) { return 1.0f / (1.0f + __expf(-x)); }

// ---------------------------------------------------------------------------
// Kernel 1: fused LSTM. Each block = 16 batch rows; 4 waves, wave w computes
// gate chunk [64w, 64w+64). Input GEMM (K=200) and recurrent GEMM (K=64) both
// via V_WMMA_F32_16X16X4_F32. h lives in LDS, c in registers.
// feat[b][h*25 + t] = relu(h_t[b][h])
// ---------------------------------------------------------------------------
__global__ __launch_bounds__(128) void lstm_kernel(
    const float* __restrict__ x,    // (B, 1, C, 5, 5)
    const float* __restrict__ Wih,  // (256, 200)
    const float* __restrict__ Whh,  // (256, 64)
    const float* __restrict__ bih,  // (256,)
    const float* __restrict__ bhh,  // (256,)
    float* __restrict__ feat)       // (B, 1600) workspace
{
    __shared__ float hlds[16 * NH];   // 16 rows x 64 : 4 KB
    __shared__ float glds[16 * NG];   // 16 rows x 256: 16 KB

    const int tid  = threadIdx.x;
    const int lane = tid & 31;
    const int gb   = (tid >> 5) * 64;      // gate-chunk base for this wave
    const int b0   = blockIdx.x * 16;      // batch tile base
    const bool hi  = lane >= 16;
    const int nn   = lane & 15;            // N col (B/D frags) / M row (A frag)
    const int ks   = hi ? 2 : 0;           // K sub-select per lane half
    const int rofs = hi ? 8 : 0;           // D row offset per lane half

    // init c (registers) and h (LDS) to zero
    float creg[8];
#pragma unroll
    for (int ii = 0; ii < 8; ++ii) {
        creg[ii] = 0.0f;
        hlds[tid + 128 * ii] = 0.0f;       // 1024 entries
    }
    __syncthreads();

    for (int t = 0; t < NT; ++t) {
        // seq[t][b][c] = x[(b*C + c)*25 + (t%5)*5 + t/5]  (the swapaxes(2,3))
        const int off = (t % 5) * 5 + (t / 5);

        // bias-initialized accumulators for 4 N-tiles (gates gb .. gb+63)
        v8f acc[4];
#pragma unroll
        for (int j = 0; j < 4; ++j) {
            const int g = gb + 16 * j + nn;
            const float bv = bih[g] + bhh[g];
#pragma unroll
            for (int r = 0; r < 8; ++r) acc[j][r] = bv;
        }

        // ---- input projection: K = 200 -> 50 WMMA K-steps ----
        const int arow = b0 + nn;
        for (int k = 0; k < 50; ++k) {
            const int kb = 4 * k + ks;
            v2f a;
            a.x = x[(arow * NC + kb    ) * NT + off];
            a.y = x[(arow * NC + kb + 1) * NT + off];
#pragma unroll
            for (int j = 0; j < 4; ++j) {
                const int g = gb + 16 * j + nn;
                v2f b;
                b.x = Wih[g * NC + kb];
                b.y = Wih[g * NC + kb + 1];
                acc[j] = WMMA_F32(a, b, acc[j]);
            }
        }

        // ---- recurrent projection: K = 64 -> 16 WMMA K-steps ----
        for (int k = 0; k < 16; ++k) {
            const int kb = 4 * k + ks;
            v2f a;
            a.x = hlds[nn * NH + kb];
            a.y = hlds[nn * NH + kb + 1];
#pragma unroll
            for (int j = 0; j < 4; ++j) {
                const int g = gb + 16 * j + nn;
                v2f b;
                b.x = Whh[g * NH + kb];
                b.y = Whh[g * NH + kb + 1];
                acc[j] = WMMA_F32(a, b, acc[j]);
            }
        }

        // spill gate tile to LDS (C/D layout: vgpr r -> rows r / r+8)
#pragma unroll
        for (int j = 0; j < 4; ++j)
#pragma unroll
            for (int r = 0; r < 8; ++r)
                glds[(r + rofs) * NG + gb + 16 * j + nn] = acc[j][r];
        __syncthreads();

        // ---- elementwise LSTM cell: 16x64 = 1024 updates over 128 threads ----
#pragma unroll
        for (int ii = 0; ii < 8; ++ii) {
            const int idx = tid + 128 * ii;
            const int row = idx >> 6;
            const int hc  = idx & 63;
            const float gi = glds[row * NG + hc];
            const float gf = glds[row * NG + 64 + hc];
            const float gg = glds[row * NG + 128 + hc];
            const float go = glds[row * NG + 192 + hc];
            const float cc = sigf(gf) * creg[ii] + sigf(gi) * tanhf(gg);
            const float hh = sigf(go) * tanhf(cc);
            creg[ii] = cc;
            hlds[row * NH + hc] = hh;  // carry (pre-ReLU)
            feat[(size_t)(b0 + row) * NF + hc * NT + t] = fmaxf(hh, 0.0f);
        }
        __syncthreads();
    }
}

// ---------------------------------------------------------------------------
// Kernel 2: per-feature batch statistics -> fused scale/shift.
// One thread per feature; consecutive threads read consecutive columns
// (coalesced along the batch sweep).
// ---------------------------------------------------------------------------
__global__ void bn_stats_kernel(const float* __restrict__ feat,
                                const float* __restrict__ gamma,
                                const float* __restrict__ beta,
                                float* __restrict__ scale,
                                float* __restrict__ shift)
{
    const int f = blockIdx.x * blockDim.x + threadIdx.x;
    if (f >= NF) return;
    float s = 0.0f, s2 = 0.0f;
    for (int b = 0; b < NB; ++b) {
        const float v = feat[(size_t)b * NF + f];
        s += v;
        s2 += v * v;
    }
    const float mu  = s * (1.0f / NB);
    const float var = s2 * (1.0f / NB) - mu * mu;   // ddof=0, matches jnp.var
    const float sc  = gamma[f] * rsqrtf(var + 1e-5f);
    scale[f] = sc;
    shift[f] = beta[f] - mu * sc;
}

// ---------------------------------------------------------------------------
// Kernel 3: out = relu(feat*scale + shift) @ fc_w^T + fc_b  via fp32 WMMA,
// one wave per 16 batch rows, K = 1600 -> 400 WMMA K-steps. Normalization +
// ReLU folded into A-fragment loads; fc_b folded into accumulator init.
// ---------------------------------------------------------------------------
__global__ __launch_bounds__(32) void fc_kernel(
    const float* __restrict__ feat,
    const float* __restrict__ scale,
    const float* __restrict__ shift,
    const float* __restrict__ fcw,   // (16, 1600)
    const float* __restrict__ fcb,   // (16,)
    float* __restrict__ out)         // (B, 16)
{
    const int lane = threadIdx.x;
    const int b0   = blockIdx.x * 16;
    const bool hi  = lane >= 16;
    const int nn   = lane & 15;
    const int ks   = hi ? 2 : 0;
    const int rofs = hi ? 8 : 0;

    v8f acc;
    const float bb = fcb[nn];
#pragma unroll
    for (int r = 0; r < 8; ++r) acc[r] = bb;

    const int arow = b0 + nn;
    for (int k = 0; k < 400; ++k) {
        const int kb = 4 * k + ks;
        v2f a, b;
        const float f0 = feat[(size_t)arow * NF + kb];
        const float f1 = feat[(size_t)arow * NF + kb + 1];
        a.x = fmaxf(f0 * scale[kb] + shift[kb], 0.0f);
        a.y = fmaxf(f1 * scale[kb + 1] + shift[kb + 1], 0.0f);
        b.x = fcw[nn * NF + kb];
        b.y = fcw[nn * NF + kb + 1];
        acc = WMMA_F32(a, b, acc);
    }

#pragma unroll
    for (int r = 0; r < 8; ++r)
        out[(b0 + r + rofs) * NCLS + nn] = acc[r];
}

// ---------------------------------------------------------------------------
extern "C" void kernel_launch(void* const* d_in, const int* in_sizes, int n_in,
                              void* d_out, int out_size, void* d_ws, size_t ws_size,
                              hipStream_t stream)
{
    const float* x     = (const float*)d_in[0];
    const float* Wih   = (const float*)d_in[1];
    const float* Whh   = (const float*)d_in[2];
    const float* bih   = (const float*)d_in[3];
    const float* bhh   = (const float*)d_in[4];
    const float* gamma = (const float*)d_in[5];
    const float* beta  = (const float*)d_in[6];
    const float* fcw   = (const float*)d_in[7];
    const float* fcb   = (const float*)d_in[8];
    float* out = (float*)d_out;

    // workspace layout: feat (4096*1600 f32) | scale (1600) | shift (1600)
    float* feat  = (float*)d_ws;
    float* scale = feat + (size_t)NB * NF;
    float* shift = scale + NF;

    lstm_kernel<<<NB / 16, 128, 0, stream>>>(x, Wih, Whh, bih, bhh, feat);
    bn_stats_kernel<<<(NF + 255) / 256, 256, 0, stream>>>(feat, gamma, beta, scale, shift);
    fc_kernel<<<NB / 16, 32, 0, stream>>>(feat, scale, shift, fcw, fcb, out);
}